// EvolutionGeneratorLognormal_27384711479788
// MI455X (gfx1250) — compile-verified
//
#include <hip/hip_runtime.h>
#include <hip/hip_bf16.h>

typedef __attribute__((ext_vector_type(2))) float v2f;
typedef __attribute__((ext_vector_type(8))) float v8f;

#define N_DATES   252
#define M_FACTORS 8
#define N_SIMS    65536
#define Z_DATE_STRIDE (M_FACTORS * N_SIMS)   // 524288 floats per date
#define COV_DATE_STRIDE (M_FACTORS * M_FACTORS)

// Each wave owns 32 consecutive sims. Per date:
//   acc(16x16 f32) += [cov ; 0] x z[sims 0..15]   (2 WMMAs, K=0..3 and 4..7)
//   acc(16x16 f32) += [0 ; cov] x z[sims 16..31]  (2 WMMAs)
// so acc VGPR r = factor r for all 32 sims (lanes 0..31 contiguous).
__global__ __launch_bounds__(256)
void evo_lognormal_kernel(const float* __restrict__ cov,
                          const float* __restrict__ var,
                          const float* __restrict__ z,
                          float* __restrict__ out) {
    const int lane       = threadIdx.x & 31;
    const int waveGlobal = blockIdx.x * (blockDim.x >> 5) + (threadIdx.x >> 5);
    const int s0         = waveGlobal * 32;           // wave-uniform
    if (s0 >= N_SIMS) return;                         // whole wave exits together

    const int  i        = lane & 15;                  // M row (A) / N col (B) within 16
    const bool hiBlock  = (i >= 8);                   // lanes feeding A-block-2
    const int  kb       = (lane >> 4) << 1;           // K base: 0 (lanes 0-15) or 2 (16-31)

    // A (cov) per-lane float offsets: row-major cov[n, row, k], row stride 8
    const int covRow = hiBlock ? (i - 8) : i;
    const int covOff = covRow * M_FACTORS + kb;       // .x=K kb, .y=K kb+1 ; +4 for K 4..7

    // B (z) per-lane float offsets: z[n, k, sim], row stride N_SIMS
    const int g0x = kb * N_SIMS + s0 + i;             // group0 sims s0..s0+15
    const int g1x = g0x + 16;                         // group1 sims s0+16..s0+31

    const float* zn = z;
    float*       on = out;

    v8f acc = {};                                     // cumsum accumulator (C/D matrix)
    const v2f zero2 = {0.0f, 0.0f};

    for (int n = 0; n < N_DATES; ++n) {
        // ---- A matrices: cov rows, dual-block packed ----
        const float* cn = cov + n * COV_DATE_STRIDE;
        v2f clo = *(const v2f*)(cn + covOff);         // K = kb, kb+1
        v2f chi = *(const v2f*)(cn + covOff + 4);     // K = kb+4, kb+5
        v2f a1lo = hiBlock ? zero2 : clo;             // cov in M rows 0..7
        v2f a1hi = hiBlock ? zero2 : chi;
        v2f a2lo = hiBlock ? clo : zero2;             // cov in M rows 8..15
        v2f a2hi = hiBlock ? chi : zero2;

        // ---- B matrices: z rows (K) x 16 sims (N) ----
        v2f b0lo, b0hi, b1lo, b1hi;
        b0lo.x = zn[g0x];
        b0lo.y = zn[g0x + 1 * N_SIMS];
        b0hi.x = zn[g0x + 4 * N_SIMS];
        b0hi.y = zn[g0x + 5 * N_SIMS];
        b1lo.x = zn[g1x];
        b1lo.y = zn[g1x + 1 * N_SIMS];
        b1hi.x = zn[g1x + 4 * N_SIMS];
        b1hi.y = zn[g1x + 5 * N_SIMS];

        // ---- 4x V_WMMA_F32_16X16X4_F32, accumulating the running cumsum ----
        acc = __builtin_amdgcn_wmma_f32_16x16x4_f32(false, a1lo, false, b0lo,
                                                    (short)0, acc, false, false);
        acc = __builtin_amdgcn_wmma_f32_16x16x4_f32(false, a1hi, false, b0hi,
                                                    (short)0, acc, false, false);
        acc = __builtin_amdgcn_wmma_f32_16x16x4_f32(false, a2lo, false, b1lo,
                                                    (short)0, acc, false, false);
        acc = __builtin_amdgcn_wmma_f32_16x16x4_f32(false, a2hi, false, b1hi,
                                                    (short)0, acc, false, false);

        // ---- bias (-var/2, uniform per factor row) + fully coalesced stores ----
        const float* vn = var + n * M_FACTORS;
#pragma unroll
        for (int r = 0; r < M_FACTORS; ++r) {
            acc[r] = fmaf(vn[r], -0.5f, acc[r]);
            on[r * N_SIMS + s0 + lane] = acc[r];
        }

        zn += Z_DATE_STRIDE;
        on += Z_DATE_STRIDE;

        // ---- stream-prefetch the z tile two dates ahead (global_prefetch_b8) ----
        if (n + 2 < N_DATES) {
            const float* zpf = zn + Z_DATE_STRIDE;    // date n+2
#pragma unroll
            for (int r = 0; r < M_FACTORS; ++r)
                __builtin_prefetch(zpf + r * N_SIMS + s0 + lane, 0, 1);
        }
    }
}

extern "C" void kernel_launch(void* const* d_in, const int* in_sizes, int n_in,
                              void* d_out, int out_size, void* d_ws, size_t ws_size,
                              hipStream_t stream) {
    const float* cov = (const float*)d_in[0];   // [252, 8, 8]
    const float* var = (const float*)d_in[1];   // [252, 8]
    const float* z   = (const float*)d_in[2];   // [252, 8, 65536]
    float*       out = (float*)d_out;           // [252, 8, 65536]

    const int waves   = N_SIMS / 32;            // 2048 waves, 32 sims each
    const int block   = 256;                    // 8 waves per block
    const int grid    = waves / (block / 32);   // 256 blocks

    evo_lognormal_kernel<<<grid, block, 0, stream>>>(cov, var, z, out);
}